// Dynamics_22041772163513
// MI455X (gfx1250) — compile-verified
//
#include <hip/hip_runtime.h>
#include <hip/hip_bf16.h>

// Lattice / integrator constants (match reference)
#define NTT 16
#define NXX 16
#define NN  256           // NT*NX
#define TSTEPS 5
#define EPSF 0.1f
// DELTA_U = 0.5  =>  x/DELTA_U == 2*x

typedef float v2f __attribute__((ext_vector_type(2)));
typedef float v8f __attribute__((ext_vector_type(8)));

// D = A*B + C on 16x16x4 f32 tiles (ISA: V_WMMA_F32_16X16X4_F32, 8-arg form)
#define WMMA_F32(a, b, c) \
  __builtin_amdgcn_wmma_f32_16x16x4_f32(false, (a), false, (b), (short)0, (c), false, false)

// ---------------------------------------------------------------------------
// Complex 16x16 matmul C = A*B entirely in LDS, one wave32, via WMMA f32.
//   Cre = Ar*Br - Ai*Bi ; Cim = Ar*Bi + Ai*Br
// A fragment (16x4, f32): lane m=lane&15, half h=lane>>4 selects K = k0+2h{,+1}
// B fragment (4x16):      lane n=lane&15, rows K = k0+2h{,+1}
// C/D (16x16):            lane n=lane&15, VGPR r -> row (h*8 + r)
// ---------------------------------------------------------------------------
__device__ __forceinline__ void cmm16(float* __restrict__ Cr, float* __restrict__ Ci,
                                      const float* __restrict__ Ar, const float* __restrict__ Ai,
                                      const float* __restrict__ Br, const float* __restrict__ Bi,
                                      int lane) {
  __syncthreads();  // producer visibility / compiler fence (single wave: ~free)
  const int m = lane & 15;
  const int h = lane >> 4;
  v8f cr, ci;
#pragma unroll
  for (int r = 0; r < 8; ++r) { cr[r] = 0.0f; ci[r] = 0.0f; }

#pragma unroll
  for (int k0 = 0; k0 < 16; k0 += 4) {
    const int ka = k0 + 2 * h;
    v2f ar, ai, br, bi, an;
    ar[0] = Ar[m * 16 + ka];       ar[1] = Ar[m * 16 + ka + 1];
    ai[0] = Ai[m * 16 + ka];       ai[1] = Ai[m * 16 + ka + 1];
    br[0] = Br[ka * 16 + m];       br[1] = Br[(ka + 1) * 16 + m];
    bi[0] = Bi[ka * 16 + m];       bi[1] = Bi[(ka + 1) * 16 + m];
    an[0] = -ai[0];                an[1] = -ai[1];   // f32 WMMA: NEG[1:0] must be 0; negate in VALU
    cr = WMMA_F32(ar, br, cr);
    cr = WMMA_F32(an, bi, cr);
    ci = WMMA_F32(ar, bi, ci);
    ci = WMMA_F32(ai, br, ci);
  }

  __syncthreads();  // WAR: everyone done reading before C is overwritten (C may be reused buffer)
  const int rb = h * 8;
#pragma unroll
  for (int r = 0; r < 8; ++r) {
    Cr[(rb + r) * 16 + m] = cr[r];
    Ci[(rb + r) * 16 + m] = ci[r];
  }
  __syncthreads();
}

// ---------------------------------------------------------------------------
// One wave32 workgroup integrates one batch row for all 5 leapfrog steps.
// ---------------------------------------------------------------------------
__global__ __launch_bounds__(32) void Dynamics_22041772163513_kernel(
    const float* __restrict__ x_in, const float* __restrict__ v_in,
    const float* __restrict__ expk, float* __restrict__ out, int B) {
  // ~82 KB LDS per workgroup (320 KB per WGP available)
  __shared__ float ek_s[NN];
  __shared__ float xs[NN], vs[NN], pc[NN], ps[NN], grd[NN];
  __shared__ float Bre[NTT * NN], Bim[NTT * NN];     // B_t blocks
  __shared__ float Gre[NTT * NN], Gim[NTT * NN];     // suffix products G'_t = B15..Bt
  __shared__ float FaR[NN], FaI[NN], FbR[NN], FbI[NN];  // prefix ping-pong
  __shared__ float WbR[NN], WbI[NN];                 // W~_t buffer
  __shared__ float QR[NN], QI[NN];                   // Q_t
  __shared__ float AugR[16 * 32], AugI[16 * 32];     // Gauss-Jordan [I+Q | W]

  const int lane = (int)threadIdx.x;
  const int b = (int)blockIdx.x;

  __builtin_prefetch(x_in + (size_t)b * NN, 0, 0);   // global_prefetch_b8
  for (int i = lane; i < NN; i += 32) {
    ek_s[i] = expk[i];
    xs[i] = x_in[(size_t)b * NN + i];
    vs[i] = v_in[(size_t)b * NN + i];
  }
  __syncthreads();

  auto compute_grad = [&]() {
    // phases: exp(i*x) per site
    for (int i = lane; i < NN; i += 32) { ps[i] = __sinf(xs[i]); pc[i] = __cosf(xs[i]); }
    __syncthreads();
    // B_t[x1,y] = ek[x1,y] * phase[t,y]
    for (int i = lane; i < NTT * NN; i += 32) {
      const int t = i >> 8, ry = i & 255, y = ry & 15;
      const float e = ek_s[ry];
      Bre[i] = e * pc[t * 16 + y];
      Bim[i] = e * ps[t * 16 + y];
    }
    __syncthreads();
    // suffix products G'_t = G'_{t+1} * B_t  (G'_15 = B_15)
    for (int i = lane; i < NN; i += 32) { Gre[15 * NN + i] = Bre[15 * NN + i]; Gim[15 * NN + i] = Bim[15 * NN + i]; }
    for (int t = 14; t >= 0; --t)
      cmm16(&Gre[t * NN], &Gim[t * NN], &Gre[(t + 1) * NN], &Gim[(t + 1) * NN],
            &Bre[t * NN], &Bim[t * NN], lane);

    float *FcR = FaR, *FcI = FaI, *FnR = FbR, *FnI = FbI;  // Fc = F'_{t-1}
    for (int t = 0; t < NTT; ++t) {
      // W~_t = product of all B except B_t = F'_{t-1} * G'_{t+1}
      const float *WR, *WI;
      if (t == 0)        { WR = &Gre[1 * NN]; WI = &Gim[1 * NN]; }
      else if (t == 15)  { WR = FcR;          WI = FcI; }
      else {
        cmm16(WbR, WbI, FcR, FcI, &Gre[(t + 1) * NN], &Gim[(t + 1) * NN], lane);
        WR = WbR; WI = WbI;
      }
      // Q_t = W~_t * B_t   (full cyclic product starting at t)
      cmm16(QR, QI, WR, WI, &Bre[t * NN], &Bim[t * NN], lane);

      // Solve (I + Q_t) S = W~_t  via complex Gauss-Jordan on [I+Q | W]
      for (int i = lane; i < 512; i += 32) {
        const int r = i >> 5, c = i & 31;
        if (c < 16) { AugR[i] = QR[r * 16 + c] + ((r == c) ? 1.0f : 0.0f); AugI[i] = QI[r * 16 + c]; }
        else        { AugR[i] = WR[r * 16 + (c - 16)];                     AugI[i] = WI[r * 16 + (c - 16)]; }
      }
      __syncthreads();
      const int c = lane;  // one augmented column per lane
      for (int k = 0; k < 16; ++k) {
        const float pr = AugR[k * 32 + k], pi = AugI[k * 32 + k];
        const float den = 1.0f / (pr * pr + pi * pi);
        const float ivr = pr * den, ivi = -pi * den;
        const float rr = AugR[k * 32 + c], rim = AugI[k * 32 + c];
        const float nr = rr * ivr - rim * ivi;
        const float ni = rr * ivi + rim * ivr;
        AugR[k * 32 + c] = nr; AugI[k * 32 + c] = ni;
        __syncthreads();
        for (int r = 0; r < 16; ++r) {
          if (r == k) continue;
          const float fr = AugR[r * 32 + k], fi = AugI[r * 32 + k];
          AugR[r * 32 + c] -= fr * nr - fi * ni;
          AugI[r * 32 + c] -= fr * ni + fi * nr;
        }
        __syncthreads();
      }
      // D_t[x] = sum_y S[x,y]*ek[y,x];  grad = 2x + 2*(sin*Dr + cos*Di)
      if (lane < 16) {
        float dr = 0.0f, di = 0.0f;
#pragma unroll
        for (int y = 0; y < 16; ++y) {
          const float e = ek_s[y * 16 + lane];
          dr += AugR[lane * 32 + 16 + y] * e;
          di += AugI[lane * 32 + 16 + y] * e;
        }
        const int idx = t * 16 + lane;
        grd[idx] = 2.0f * xs[idx] + 2.0f * (ps[idx] * dr + pc[idx] * di);
      }
      __syncthreads();
      // advance prefix: F'_t = B_t * F'_{t-1}
      if (t == 0) {
        for (int i = lane; i < NN; i += 32) { FcR[i] = Bre[i]; FcI[i] = Bim[i]; }
        __syncthreads();
      } else if (t < 15) {
        cmm16(FnR, FnI, &Bre[t * NN], &Bim[t * NN], FcR, FcI, lane);
        float* s;
        s = FcR; FcR = FnR; FnR = s;
        s = FcI; FcI = FnI; FnI = s;
      }
    }
  };

  // Leapfrog: v_h = v - eps/2 * g(x); x += eps*v_h; v = v_h - eps/2 * g(x)
  for (int step = 0; step < TSTEPS; ++step) {
    compute_grad();
    for (int i = lane; i < NN; i += 32) {
      const float vh = vs[i] - 0.5f * EPSF * grd[i];
      xs[i] += EPSF * vh;
      vs[i] = vh;
    }
    __syncthreads();
    compute_grad();
    for (int i = lane; i < NN; i += 32) vs[i] -= 0.5f * EPSF * grd[i];
    __syncthreads();
  }

  // outputs: [x (B*256) | v (B*256) | j (B)]
  float* out_x = out;
  float* out_v = out + (size_t)B * NN;
  float* out_j = out + (size_t)2 * B * NN;
  for (int i = lane; i < NN; i += 32) {
    out_x[(size_t)b * NN + i] = xs[i];
    out_v[(size_t)b * NN + i] = vs[i];
  }
  if (lane == 0) out_j[b] = 0.0f;
}

extern "C" void kernel_launch(void* const* d_in, const int* in_sizes, int n_in,
                              void* d_out, int out_size, void* d_ws, size_t ws_size,
                              hipStream_t stream) {
  const float* x  = (const float*)d_in[0];
  const float* v  = (const float*)d_in[1];
  const float* ek = (const float*)d_in[2];
  const int B = in_sizes[0] / NN;  // 128
  Dynamics_22041772163513_kernel<<<dim3(B), dim3(32), 0, stream>>>(x, v, ek, (float*)d_out, B);
}